// VQVAE_70540542869934
// MI455X (gfx1250) — compile-verified
//
#include <hip/hip_runtime.h>
#include <hip/hip_bf16.h>

typedef float v2f __attribute__((ext_vector_type(2)));
typedef float v8f __attribute__((ext_vector_type(8)));

#define KCODES 8192
#define CDIM 64
#define NQ 8192          // B*H*W = 8*32*32
#define HWSZ 1024        // H*W
#define LDS_STRIDE 68    // 64 + 4 floats pad -> conflict-free v2f reads

// ---------------------------------------------------------------------------
// Kernel 0: e2[k] = |codebook[k]|^2 for all 8192 codes (tiny, L2-resident).
// ---------------------------------------------------------------------------
__global__ __launch_bounds__(256) void vq_e2_kernel(
    const float* __restrict__ codebook, float* __restrict__ e2g)
{
  const int k = blockIdx.x * 256 + threadIdx.x;
  const float4* row = (const float4*)(codebook + (size_t)k * CDIM);
  float s = 0.f;
  #pragma unroll
  for (int i = 0; i < CDIM / 4; ++i) {
    const float4 v = row[i];
    s += v.x * v.x + v.y * v.y + v.z * v.z + v.w * v.w;
  }
  e2g[k] = s;
}

// ---------------------------------------------------------------------------
// Kernel 1: fused fp32 GEMM (x @ codebook^T via V_WMMA_F32_16X16X4_F32) +
// row-wise argmin of (|e|^2 - 2*xe). |x|^2 is row-constant and dropped
// (argmin-invariant). One wave = 16 queries; one block = 8 waves = 128
// queries sharing double-buffered LDS-staged codebook tiles.
// sched_group_barrier clusters the 8 DS-reads ahead of the 16-WMMA burst so
// the scheduler uses progressive dscnt waits instead of 8 full drains.
// ---------------------------------------------------------------------------
__global__ __launch_bounds__(256) void vq_argmin_kernel(
    const float* __restrict__ x, const float* __restrict__ codebook,
    const float* __restrict__ e2g,
    int* __restrict__ idx_ws, float* __restrict__ idx_out)
{
  __shared__ float lds[2 * 16 * LDS_STRIDE];

  const int tid  = threadIdx.x;
  const int wave = tid >> 5;
  const int lane = tid & 31;
  const int col  = lane & 15;   // N (or M for A-frag) index within tile
  const int hi   = lane >> 4;   // half-wave select
  const int m_base = blockIdx.x * 128 + wave * 16;

  // --- Load this wave's 16 query vectors into A fragments (strided over C) ---
  // x layout: [B, C, H, W]; query q -> x[b*C*HW + c*HW + hw]
  const int q = m_base + col;
  const float* xq = x + (size_t)(q >> 10) * (CDIM * HWSZ) + (q & (HWSZ - 1));
  v2f a[16];
  #pragma unroll
  for (int ks = 0; ks < 16; ++ks) {
    const int k0 = 4 * ks + 2 * hi;                 // A frag: K = 4*ks + 2*hi + v
    a[ks].x = xq[(size_t)k0 * HWSZ];
    a[ks].y = xq[(size_t)(k0 + 1) * HWSZ];
  }

  float best[8];
  int   bidx[8];
  #pragma unroll
  for (int r = 0; r < 8; ++r) { best[r] = 3.4e38f; bidx[r] = 0; }

  // --- Cooperative tile staging: 16 codes x 64 floats = 4KB = 256 x float4 ---
  const int lrow  = tid >> 4;          // 0..15  (code row within tile)
  const int lcol4 = (tid & 15) << 2;   // 0..60  (float4 column)
  {
    const float4 v = *(const float4*)(codebook + (size_t)lrow * CDIM + lcol4);
    *(float4*)&lds[lrow * LDS_STRIDE + lcol4] = v;
  }
  __syncthreads();

  const int NTILES = KCODES / 16;
  for (int t = 0; t < NTILES; ++t) {
    // Register-staged prefetch of next tile (hides L2 latency behind WMMA).
    float4 stage = make_float4(0.f, 0.f, 0.f, 0.f);
    if (t + 1 < NTILES) {
      stage = *(const float4*)(codebook +
                ((size_t)(t + 1) * 16 + lrow) * CDIM + lcol4);
    }
    const int   nidx = t * 16 + col;
    const float e2n  = e2g[nidx];     // precomputed |E[n]|^2, coalesced dword

    const float* buf = &lds[(t & 1) * (16 * LDS_STRIDE)];

    // Bulk-load all B fragments (clustered DS reads), then WMMA burst.
    v2f bfrag[16];
    #pragma unroll
    for (int ks = 0; ks < 16; ++ks) {
      // B frag: lane holds E[n][k] with n = col, k = 4*ks + 2*hi + v
      bfrag[ks] = *(const v2f*)&buf[col * LDS_STRIDE + 4 * ks + 2 * hi];
    }

    v8f acc = {0.f, 0.f, 0.f, 0.f, 0.f, 0.f, 0.f, 0.f};
    #pragma unroll
    for (int ks = 0; ks < 16; ++ks) {
      acc = __builtin_amdgcn_wmma_f32_16x16x4_f32(
              false, a[ks], false, bfrag[ks], (short)0, acc, false, false);
    }

    // Scheduling: group the 8 ds_load_2addr_b64 first, then the 16 WMMAs.
    __builtin_amdgcn_sched_group_barrier(0x100, 8, 0);   // DS reads
    __builtin_amdgcn_sched_group_barrier(0x008, 16, 0);  // matrix ops

    #pragma unroll
    for (int r = 0; r < 8; ++r) {
      // D layout: row m = r + 8*hi, col n = lane&15
      const float d2 = e2n - 2.0f * acc[r];   // + |x_m|^2 (row-constant, dropped)
      if (d2 < best[r]) { best[r] = d2; bidx[r] = nidx; }
    }

    if (t + 1 < NTILES) {
      *(float4*)&lds[((t + 1) & 1) * (16 * LDS_STRIDE) +
                     lrow * LDS_STRIDE + lcol4] = stage;
    }
    __syncthreads();
  }

  // --- Argmin across the 16 lanes of each half-wave (index tie-break: min) ---
  #pragma unroll
  for (int r = 0; r < 8; ++r) {
    float bv = best[r]; int bi = bidx[r];
    #pragma unroll
    for (int mask = 8; mask >= 1; mask >>= 1) {
      const float ov = __shfl_xor(bv, mask, 32);
      const int   oi = __shfl_xor(bi, mask, 32);
      if (ov < bv || (ov == bv && oi < bi)) { bv = ov; bi = oi; }
    }
    if (col == 0) {
      const int m = r + 8 * hi;
      idx_ws[m_base + m]  = bi;
      idx_out[m_base + m] = (float)bi;
    }
  }
}

// ---------------------------------------------------------------------------
// Kernel 2: gather codebook[idx], write NCHW quant_out, per-query MSE partial.
// One 64-thread block per query (c = threadIdx.x). Deterministic (no atomics).
// ---------------------------------------------------------------------------
__global__ __launch_bounds__(64) void vq_gather_kernel(
    const float* __restrict__ x, const float* __restrict__ codebook,
    const int* __restrict__ idx_ws, float* __restrict__ quant_out,
    float* __restrict__ partial)
{
  const int q = blockIdx.x;
  const int c = threadIdx.x;
  const int code = idx_ws[q];
  const int b  = q >> 10;
  const int hw = q & (HWSZ - 1);

  const size_t off = (size_t)b * (CDIM * HWSZ) + (size_t)c * HWSZ + hw;
  const float ev = codebook[(size_t)code * CDIM + c];
  const float xv = x[off];
  quant_out[off] = ev;              // straight-through value == quant

  float d = ev - xv;
  float s = d * d;
  #pragma unroll
  for (int mask = 16; mask >= 1; mask >>= 1) s += __shfl_xor(s, mask, 32);

  __shared__ float ws2[2];
  if ((threadIdx.x & 31) == 0) ws2[threadIdx.x >> 5] = s;
  __syncthreads();
  if (threadIdx.x == 0) partial[q] = ws2[0] + ws2[1];
}

// ---------------------------------------------------------------------------
// Kernel 3: reduce per-query partials to the (identical) loss scalars.
// ---------------------------------------------------------------------------
__global__ __launch_bounds__(256) void vq_loss_kernel(
    const float* __restrict__ partial, float* __restrict__ out_scalars)
{
  __shared__ float red[256];
  float s = 0.f;
  for (int i = threadIdx.x; i < NQ; i += 256) s += partial[i];
  red[threadIdx.x] = s;
  __syncthreads();
  #pragma unroll
  for (int step = 128; step >= 1; step >>= 1) {
    if ((int)threadIdx.x < step) red[threadIdx.x] += red[threadIdx.x + step];
    __syncthreads();
  }
  if (threadIdx.x == 0) {
    const float loss = red[0] / (float)((size_t)NQ * CDIM);
    out_scalars[0] = loss;   // commitment_loss
    out_scalars[1] = loss;   // codebook_loss (same value)
  }
}

// ---------------------------------------------------------------------------
extern "C" void kernel_launch(void* const* d_in, const int* in_sizes, int n_in,
                              void* d_out, int out_size, void* d_ws, size_t ws_size,
                              hipStream_t stream) {
  const float* x        = (const float*)d_in[0];   // [8,64,32,32] fp32
  const float* codebook = (const float*)d_in[1];   // [8192,64] fp32
  float* out = (float*)d_out;

  // Output layout (flattened, return order): quant_out[524288], commit[1],
  // codebook_loss[1], indices[8192] (as float).
  float* quant_out = out;
  float* scalars   = out + (size_t)NQ * CDIM;         // 524288
  float* idx_out   = out + (size_t)NQ * CDIM + 2;     // 524290

  int*   idx_ws  = (int*)d_ws;                                       // 32 KB
  float* partial = (float*)((char*)d_ws + (size_t)NQ * sizeof(int)); // 32 KB
  float* e2g     = partial + NQ;                                     // 32 KB

  vq_e2_kernel<<<KCODES / 256, 256, 0, stream>>>(codebook, e2g);
  vq_argmin_kernel<<<NQ / 128, 256, 0, stream>>>(x, codebook, e2g, idx_ws, idx_out);
  vq_gather_kernel<<<NQ, 64, 0, stream>>>(x, codebook, idx_ws, quant_out, partial);
  vq_loss_kernel<<<1, 256, 0, stream>>>(partial, scalars);
}